// VoxelPooling_790273982604
// MI455X (gfx1250) — compile-verified
//
#include <hip/hip_runtime.h>
#include <stdint.h>

// VoxelPooling (LSS BEV pooling), MI455X / gfx1250.
//
// Roofline: reads 10.8MB geom + 288MB features (streamed once -> NT loads),
// 72M f32 scatter-atomics into a 168MB output that fits in the 192MB L2.
// HBM floor ~20us @ 23.3TB/s; real limiter is L2 atomic rate, so all atomics
// are non-returning global_atomic_add_f32 (STOREcnt, no wait-for-return).
// Geometry is staged per-block through LDS with the CDNA5 async
// global->LDS DMA path (ASYNCcnt) in fully coalesced 16B/lane chunks.

namespace {
constexpr int kB  = 8;
constexpr int kD  = 40;
constexpr int kH  = 32;
constexpr int kW  = 88;
constexpr int kC  = 80;
constexpr int kNX = 256;
constexpr int kNY = 256;
constexpr int kHW  = kH * kW;    // 2816
constexpr int kNXY = kNX * kNY;  // 65536
constexpr int TPB  = 256;        // 8 waves (wave32)
}  // namespace

__global__ __launch_bounds__(TPB) void voxel_pool_scatter(
    const float* __restrict__ geom,   // (N, 3)
    const float* __restrict__ x,      // (B, D, C, H, W)
    float* __restrict__ out) {        // (B, C, NX, NY), pre-zeroed
  __shared__ float sg[TPB * 3];       // 3 KB of geometry per block

  const int t = threadIdx.x;
  const int block_start = blockIdx.x * TPB;

  // ---- Async-stage 256 geometry rows (3072 B) into LDS: 192 lanes x B128.
  {
    const unsigned off = (unsigned)t * 16u;
    if (off < (unsigned)(TPB * 3 * 4)) {
      // Generic pointers to LDS carry the LDS byte offset in their low 32 bits
      // (shared aperture lives entirely in addr[63:32] on CDNA5).
      unsigned lds_addr =
          (unsigned)(unsigned long long)(uintptr_t)(&sg[0]) + off;
      unsigned long long gaddr = (unsigned long long)(uintptr_t)geom +
                                 (unsigned long long)block_start * 12ull + off;
      asm volatile("global_load_async_to_lds_b128 %0, %1, off"
                   :: "v"(lds_addr), "v"(gaddr)
                   : "memory");
    }
  }
  asm volatile("s_wait_asynccnt 0" ::: "memory");
  __syncthreads();

  const float gx = sg[t * 3 + 0];     // stride-3 -> bank-conflict-free
  const float gy = sg[t * 3 + 1];
  const float gz = sg[t * 3 + 2];

  // Bounds filter exactly as the reference (strict upper bound).
  const bool keep = (gx >= -51.2f) && (gx < 51.2f) &&
                    (gy >= -51.2f) && (gy < 51.2f) &&
                    (gz >= -10.0f) && (gz <  10.0f);
  if (!keep) return;

  // IEEE f32 division to match jnp's ((g - lo) / dd) rounding, then trunc+clamp.
  int ix = (int)((gx + 51.2f) / 0.4f);
  int iy = (int)((gy + 51.2f) / 0.4f);
  ix = min(max(ix, 0), kNX - 1);
  iy = min(max(iy, 0), kNY - 1);

  const int n = block_start + t;
  const int w = n % kW;
  int r = n / kW;
  const int h = r % kH; r /= kH;
  const int d = r % kD;
  const int b = r / kD;

  // x[b, d, c, h, w]: consecutive lanes -> consecutive w -> coalesced loads.
  const float* __restrict__ xp = x + ((b * kD + d) * kC) * kHW + h * kW + w;
  // out[b, c, ix, iy]: per-channel stride 256KB spreads atomics over L2 banks.
  float* __restrict__ op = out + b * kC * kNXY + ix * kNY + iy;

  #pragma unroll 8
  for (int c = 0; c < kC; ++c) {
    // NT: features are read exactly once -- keep L2 free for the atomic target.
    const float v = __builtin_nontemporal_load(xp + c * kHW);
    unsafeAtomicAdd(op + c * kNXY, v);  // non-returning global_atomic_add_f32
  }
}

extern "C" void kernel_launch(void* const* d_in, const int* in_sizes, int n_in,
                              void* d_out, int out_size, void* d_ws, size_t ws_size,
                              hipStream_t stream) {
  const float* geom = (const float*)d_in[0];  // (B,D,H,W,3) f32, flat (N,3)
  const float* x    = (const float*)d_in[1];  // (B,D,C,H,W) f32
  float* out        = (float*)d_out;          // (B, C, NX, NY) f32

  // Output must be re-zeroed every call (harness poisons it once, never restores).
  hipMemsetAsync(d_out, 0, (size_t)out_size * sizeof(float), stream);

  const int npts   = in_sizes[0] / 3;            // 901120
  const int blocks = (npts + TPB - 1) / TPB;     // 3520, no tail
  voxel_pool_scatter<<<blocks, TPB, 0, stream>>>(geom, x, out);

  (void)n_in; (void)d_ws; (void)ws_size;
}